// DETECTiveGNN_18253611008270
// MI455X (gfx1250) — compile-verified
//
#include <hip/hip_runtime.h>
#include <hip/hip_bf16.h>

typedef __attribute__((ext_vector_type(2))) float v2f;
typedef __attribute__((ext_vector_type(8))) float v8f;

#define HDIM 64
#define INDIM 128
#define NEG_SLOPE 0.2f

// Monotone order-preserving float<->uint encoding for atomicMax over floats
// (handles negative values from leaky_relu).
__device__ __forceinline__ unsigned f2ord(float f) {
  unsigned u = __float_as_uint(f);
  return (u & 0x80000000u) ? ~u : (u | 0x80000000u);
}
__device__ __forceinline__ float ord2f(unsigned u) {
  unsigned b = (u & 0x80000000u) ? (u & 0x7fffffffu) : ~u;
  return __uint_as_float(b);
}

__global__ void zero_kernel(float* __restrict__ p, long long n) {
  long long i = (long long)blockIdx.x * blockDim.x + threadIdx.x;
  long long stride = (long long)gridDim.x * blockDim.x;
  for (; i < n; i += stride) p[i] = 0.0f;
}

// C[rows,64] = A[rows,kdim] @ W[kdim,64] using fp32 WMMA 16x16x4.
// One wave per 16x16 output tile. rows is a multiple of 16 for this problem
// (N=100000 = 6250*16); ntile in [0,4) covers the 64 output columns.
__global__ __launch_bounds__(256)
void gemm_wmma_f32(const float* __restrict__ A, const float* __restrict__ W,
                   float* __restrict__ C, int rows, int kdim) {
  int wave = (int)(((long long)blockIdx.x * blockDim.x + threadIdx.x) >> 5);
  int lane = threadIdx.x & 31;
  int mtile = wave >> 2;
  int m0 = mtile * 16;
  if (m0 >= rows) return;              // wave-uniform: EXEC all-ones for WMMA
  int n0 = (wave & 3) * 16;
  int lh = lane >> 4;                   // half-wave select (K pairs)
  int li = lane & 15;                   // row (A) / col (B,C)

  const float* arow = A + (size_t)(m0 + li) * kdim + lh * 2;
  const float* wcol = W + (size_t)(lh * 2) * HDIM + n0 + li;

  v8f acc = {0.f, 0.f, 0.f, 0.f, 0.f, 0.f, 0.f, 0.f};
  for (int k = 0; k < kdim; k += 4) {
    // A fragment 16x4: lanes0-15 hold K=lh*2+{0,1} for row li  (8B aligned)
    v2f a = *(const v2f*)(arow + k);
    // B fragment 4x16: lane li = column, VGPR pair = K=lh*2+{0,1}
    v2f b;
    b.x = wcol[k * HDIM];
    b.y = wcol[(k + 1) * HDIM];
    acc = __builtin_amdgcn_wmma_f32_16x16x4_f32(false, a, false, b,
                                                (short)0, acc, false, false);
  }
  // C 16x16 f32: lane li = col, VGPR v = row lh*8+v
  float* crow = C + (size_t)(m0 + lh * 8) * HDIM + n0 + li;
#pragma unroll
  for (int v = 0; v < 8; ++v) crow[(size_t)v * HDIM] = acc[v];
}

// alpha_src[i] = h[i,:].att_src ; alpha_dst[i] = h[i,:].att_dst  (wave per node)
__global__ void attn_kernel(const float* __restrict__ h,
                            const float* __restrict__ att_s,
                            const float* __restrict__ att_d,
                            float* __restrict__ as, float* __restrict__ ad,
                            int n) {
  long long wid = ((long long)blockIdx.x * blockDim.x + threadIdx.x) >> 5;
  int lane = threadIdx.x & 31;
  if (wid >= n) return;
  const float* row = h + (size_t)wid * HDIM;
  float vs = row[lane] * att_s[lane] + row[lane + 32] * att_s[lane + 32];
  float vd = row[lane] * att_d[lane] + row[lane + 32] * att_d[lane + 32];
#pragma unroll
  for (int off = 16; off; off >>= 1) {
    vs += __shfl_xor(vs, off, 32);
    vd += __shfl_xor(vd, off, 32);
  }
  if (lane == 0) { as[wid] = vs; ad[wid] = vd; }
}

__device__ __forceinline__ void edge_src_dst(const int* __restrict__ ei,
                                             long long e, int E,
                                             int& src, int& dst) {
  if (e < E) { src = ei[e]; dst = ei[(long long)E + e]; }
  else       { src = dst = (int)(e - E); }
}

// Pass A: segment max of leaky_relu scores (encoded atomicMax) + degree count
__global__ void edge_max_deg(const int* __restrict__ ei, int E, int n,
                             const float* __restrict__ as,
                             const float* __restrict__ ad,
                             unsigned* __restrict__ menc,
                             float* __restrict__ deg) {
  long long e = (long long)blockIdx.x * blockDim.x + threadIdx.x;
  long long ET = (long long)E + n;
  if (e >= ET) return;
  int src, dst; edge_src_dst(ei, e, E, src, dst);
  float t = as[src] + ad[dst];
  float ev = t > 0.f ? t : NEG_SLOPE * t;
  atomicMax(&menc[dst], f2ord(ev));
  atomicAdd(&deg[dst], 1.0f);
}

// Pass B: ex = exp(e - m[dst]) stored per-edge; segment sum of ex
__global__ void edge_exp(const int* __restrict__ ei, int E, int n,
                         const float* __restrict__ as,
                         const float* __restrict__ ad,
                         const unsigned* __restrict__ menc,
                         float* __restrict__ exb, float* __restrict__ ssum) {
  long long e = (long long)blockIdx.x * blockDim.x + threadIdx.x;
  long long ET = (long long)E + n;
  if (e >= ET) return;
  int src, dst; edge_src_dst(ei, e, E, src, dst);
  float t = as[src] + ad[dst];
  float ev = t > 0.f ? t : NEG_SLOPE * t;
  float x = __expf(ev - ord2f(menc[dst]));
  exb[e] = x;
  atomicAdd(&ssum[dst], x);
}

// Pass C: out[dst,:] += (ex/s[dst]) * h[src,:]   (one wave per edge)
__global__ void gat_scatter(const int* __restrict__ ei, int E, int n,
                            const float* __restrict__ exb,
                            const float* __restrict__ ssum,
                            const float* __restrict__ h,
                            float* __restrict__ out) {
  long long wid = ((long long)blockIdx.x * blockDim.x + threadIdx.x) >> 5;
  int lane = threadIdx.x & 31;
  long long ET = (long long)E + n;
  if (wid >= ET) return;
  int src, dst; edge_src_dst(ei, wid, E, src, dst);
  float alpha = exb[wid] / (ssum[dst] + 1e-16f);
  const float* hr = h + (size_t)src * HDIM;
  float* orow = out + (size_t)dst * HDIM;
  atomicAdd(&orow[lane], alpha * hr[lane]);
  atomicAdd(&orow[lane + 32], alpha * hr[lane + 32]);
}

// GCN scatter: out[dst,:] += dinv[src]*dinv[dst] * h[src,:]  (wave per edge)
__global__ void gcn_scatter(const int* __restrict__ ei, int E, int n,
                            const float* __restrict__ dinv,
                            const float* __restrict__ h,
                            float* __restrict__ out) {
  long long wid = ((long long)blockIdx.x * blockDim.x + threadIdx.x) >> 5;
  int lane = threadIdx.x & 31;
  long long ET = (long long)E + n;
  if (wid >= ET) return;
  int src, dst; edge_src_dst(ei, wid, E, src, dst);
  float norm = dinv[src] * dinv[dst];
  const float* hr = h + (size_t)src * HDIM;
  float* orow = out + (size_t)dst * HDIM;
  atomicAdd(&orow[lane], norm * hr[lane]);
  atomicAdd(&orow[lane + 32], norm * hr[lane + 32]);
}

// outp = (acc + bias) optionally relu'd
__global__ void bias_act(const float* __restrict__ acc,
                         const float* __restrict__ b,
                         float* __restrict__ outp, long long total, int relu) {
  long long i = (long long)blockIdx.x * blockDim.x + threadIdx.x;
  long long stride = (long long)gridDim.x * blockDim.x;
  for (; i < total; i += stride) {
    float v = acc[i] + b[i & (HDIM - 1)];
    outp[i] = relu ? fmaxf(v, 0.f) : v;
  }
}

__global__ void dinv_kernel(float* __restrict__ deg, int n) {
  int i = blockIdx.x * blockDim.x + threadIdx.x;
  if (i >= n) return;
  float d = deg[i];
  deg[i] = d > 0.f ? rsqrtf(d) : 0.f;
}

extern "C" void kernel_launch(void* const* d_in, const int* in_sizes, int n_in,
                              void* d_out, int out_size, void* d_ws, size_t ws_size,
                              hipStream_t stream) {
  const float* x       = (const float*)d_in[0];
  const int*   ei      = (const int*)d_in[1];
  const float* W_gat   = (const float*)d_in[2];
  const float* att_src = (const float*)d_in[3];
  const float* att_dst = (const float*)d_in[4];
  const float* b_gat   = (const float*)d_in[5];
  const float* W1      = (const float*)d_in[6];
  const float* b1      = (const float*)d_in[7];
  const float* W2      = (const float*)d_in[8];
  const float* b2      = (const float*)d_in[9];

  int N = in_sizes[0] / INDIM;
  int E = in_sizes[1] / 2;
  long long ET = (long long)E + N;
  long long NH = (long long)N * HDIM;

  float* ws = (float*)d_ws;
  float*    bufA = ws;                                 // N*64
  float*    bufB = bufA + NH;                          // N*64
  unsigned* menc = (unsigned*)(bufB + NH);             // N   (zeroed)
  float*    ssum = (float*)(menc + N);                 // N   (zeroed)
  float*    deg  = ssum + N;                           // N   (zeroed)
  float*    asrc = deg + N;                            // N
  float*    adst = asrc + N;                           // N
  float*    exb  = adst + N;                           // E+N
  float*    out  = (float*)d_out;

  dim3 blk(256);
  int zgrid = 2048;
  int egrid = (int)((ET + 255) / 256);                 // thread per edge
  int wgrid = (int)((ET * 32 + 255) / 256);            // wave per edge
  int pgrid = 2048;                                    // pointwise grid-stride
  int gemm_blocks = (int)((((long long)((N + 15) / 16) * 4) * 32 + 255) / 256);
  int agrid = (int)(((long long)N * 32 + 255) / 256);  // wave per node

  // ---- GAT layer ----
  // zero: bufB (accumulator) + menc + ssum + deg (contiguous span)
  zero_kernel<<<zgrid, blk, 0, stream>>>(bufB, NH + 3LL * N);
  gemm_wmma_f32<<<gemm_blocks, blk, 0, stream>>>(x, W_gat, bufA, N, INDIM);
  attn_kernel<<<agrid, blk, 0, stream>>>(bufA, att_src, att_dst, asrc, adst, N);
  edge_max_deg<<<egrid, blk, 0, stream>>>(ei, E, N, asrc, adst, menc, deg);
  edge_exp<<<egrid, blk, 0, stream>>>(ei, E, N, asrc, adst, menc, exb, ssum);
  gat_scatter<<<wgrid, blk, 0, stream>>>(ei, E, N, exb, ssum, bufA, bufB);
  bias_act<<<pgrid, blk, 0, stream>>>(bufB, b_gat, bufA, NH, 1);  // bufA = h1

  // degree -> dinv (shared by both GCN layers)
  dinv_kernel<<<(N + 255) / 256, blk, 0, stream>>>(deg, N);

  // ---- GCN layer 1 ----
  gemm_wmma_f32<<<gemm_blocks, blk, 0, stream>>>(bufA, W1, bufB, N, HDIM);
  zero_kernel<<<zgrid, blk, 0, stream>>>(bufA, NH);
  gcn_scatter<<<wgrid, blk, 0, stream>>>(ei, E, N, deg, bufB, bufA);
  bias_act<<<pgrid, blk, 0, stream>>>(bufA, b1, bufB, NH, 1);     // bufB = h2

  // ---- GCN layer 2 ----
  gemm_wmma_f32<<<gemm_blocks, blk, 0, stream>>>(bufB, W2, bufA, N, HDIM);
  zero_kernel<<<zgrid, blk, 0, stream>>>(out, NH);
  gcn_scatter<<<wgrid, blk, 0, stream>>>(ei, E, N, deg, bufA, out);
  bias_act<<<pgrid, blk, 0, stream>>>(out, b2, out, NH, 0);       // final (no relu)
}